// MeshNN_59227599012230
// MI455X (gfx1250) — compile-verified
//
#include <hip/hip_runtime.h>

// MI455X / gfx1250, wave32. f32 GEMV of on-the-fly hat-function matrix via
// V_WMMA_F32_16X16X4_F32 (full f32 precision matrix op), node params staged in LDS.

typedef __attribute__((ext_vector_type(2))) float v2f;
typedef __attribute__((ext_vector_type(8))) float v8f;

#define NPMAX 512        // reference NP
#define ROWS_PER_BLOCK 128  // 8 waves * 16 rows

__device__ __forceinline__ float hatfn(float x, float xi, float invL, float invR) {
    // relu(1 - relu(xi-x)*invL) + relu(1 - relu(x-xi)*invR) - 1
    float t = fmaxf(1.0f - fmaxf(xi - x, 0.0f) * invL, 0.0f);
    float b = fmaxf(1.0f - fmaxf(x - xi, 0.0f) * invR, 0.0f);
    return t + b - 1.0f;
}

__global__ __launch_bounds__(256) void meshnn_wmma_kernel(
    const float* __restrict__ x,
    const float* __restrict__ coords,
    const float* __restrict__ w_uu,
    const float* __restrict__ w_dd,
    float* __restrict__ out,
    int n, int np)
{
    const float TP = 1.0f - 1.0f / 150.0f;
    const float TM = 1.0f + 1.0f / 150.0f;

    __shared__ float  sc[NPMAX];   // coords
    __shared__ float4 sp[NPMAX];   // per interior node k: {xi, 1/(xi-c[k]), 1/(c[k+2]-xi), w_uu[k]}

    const int tid  = threadIdx.x;
    int nInt = np - 2;
    if (nInt > NPMAX) nInt = NPMAX;

    for (int i = tid; i < np && i < NPMAX; i += blockDim.x) sc[i] = coords[i];
    __syncthreads();

    // Precompute clamped apex + edge reciprocals per interior node; pad with zeros
    // (hat evaluates to finite 1.0, weight 0 -> exact zero contribution).
    for (int k = tid; k < NPMAX; k += blockDim.x) {
        float4 p = make_float4(0.0f, 0.0f, 0.0f, 0.0f);
        if (k < nInt) {
            float cim1 = sc[k], ci = sc[k + 1], cip1 = sc[k + 2];
            float xi = fmaxf(fminf(ci, TP * cip1), TM * cim1);
            p.x = xi;
            p.y = 1.0f / (xi - cim1);
            p.z = 1.0f / (cip1 - xi);
            p.w = w_uu[k];
        }
        sp[k] = p;
    }
    __syncthreads();

    // Boundary hat parameters (uniform scalars).
    const float c0  = sc[0], c1 = sc[1];
    const float cL2 = sc[np - 2], cL1 = sc[np - 1];
    const float e0  = c0 - cL1 * 0.01f;          // coords[0] - coords[-1]/100
    const float xi0 = fmaxf(fminf(c0, TP * c1), TM * e0);
    const float i0L = 1.0f / (xi0 - e0),  i0R = 1.0f / (c1 - xi0);
    const float eL  = cL1 * 1.01f;               // coords[-1]*(1 + 1/100)
    const float xiL = fmaxf(fminf(cL1, TP * eL), TM * cL2);
    const float iLL = 1.0f / (xiL - cL2), iLR = 1.0f / (eL - xiL);
    const float wd0 = w_dd[0], wd1 = w_dd[1];

    const int wave = tid >> 5;          // 0..7
    const int lane = tid & 31;
    const int m    = lane & 15;         // row within 16-row tile (both half-waves)
    const int half = lane >> 4;         // 0: K={0,1}, 1: K={2,3} of each step

    const long base = (long)blockIdx.x * ROWS_PER_BLOCK + wave * 16;
    const float xm = (base + m < n) ? x[base + m] : 0.0f;

    // Boundary contribution for this lane's row.
    const float udd = wd0 * hatfn(xm, xi0, i0L, i0R) + wd1 * hatfn(xm, xiL, iLL, iLR);

    // GEMV: D[m][0] += sum_k phi(x_m, node k) * w_uu[k], K in chunks of 4 via f32 WMMA.
    v8f acc = {};
    const int kPad = (nInt + 3) & ~3;
    for (int k0 = 0; k0 < kPad; k0 += 4) {
        const int kb = k0 + 2 * half;          // A layout: lanes<16 -> K=k0,k0+1; lanes>=16 -> K=k0+2,k0+3
        float4 p0 = sp[kb];
        float4 p1 = sp[kb + 1];
        v2f a, b;
        a.x = hatfn(xm, p0.x, p0.y, p0.z);     // A VGPR0
        a.y = hatfn(xm, p1.x, p1.y, p1.z);     // A VGPR1
        // B 4x16: nonzero only in column 0 -> lanes 0 and 16.
        b.x = (m == 0) ? p0.w : 0.0f;          // B VGPR0: rows K=0 (lane0) / K=2 (lane16)
        b.y = (m == 0) ? p1.w : 0.0f;          // B VGPR1: rows K=1 (lane0) / K=3 (lane16)
        acc = __builtin_amdgcn_wmma_f32_16x16x4_f32(
            /*neg_a=*/false, a, /*neg_b=*/false, b,
            /*c_mod=*/(short)0, acc, /*reuse_a=*/false, /*reuse_b=*/false);
    }

    // Column 0 of D: lane 0 holds rows 0..7 (acc[0..7]), lane 16 holds rows 8..15.
    // Merge per-row boundary term (lives in lane r / lane 8+r) via wave32 shuffles.
#pragma unroll
    for (int r = 0; r < 8; ++r) {
        float d0 = __shfl(udd, r, 32);
        float d1 = __shfl(udd, 8 + r, 32);
        float v = acc[r] + (half ? d1 : d0);
        if (lane == 0  && base + r     < n) out[base + r]     = v;
        if (lane == 16 && base + 8 + r < n) out[base + 8 + r] = v;
    }
}

extern "C" void kernel_launch(void* const* d_in, const int* in_sizes, int n_in,
                              void* d_out, int out_size, void* d_ws, size_t ws_size,
                              hipStream_t stream) {
    const float* x      = (const float*)d_in[0];
    const float* coords = (const float*)d_in[1];
    const float* w_uu   = (const float*)d_in[2];
    const float* w_dd   = (const float*)d_in[3];
    float* out = (float*)d_out;

    const int n  = in_sizes[0];   // 131072
    const int np = in_sizes[1];   // 512

    const int blocks = (n + ROWS_PER_BLOCK - 1) / ROWS_PER_BLOCK;
    meshnn_wmma_kernel<<<blocks, 256, 0, stream>>>(x, coords, w_uu, w_dd, out, n, np);
}